// RecylingEmbedder_42717744726533
// MI455X (gfx1250) — compile-verified
//
#include <hip/hip_runtime.h>

// ---------------------------------------------------------------------------
// RecyclingEmbedder forward (AlphaFold-style), MI455X / gfx1250, wave32.
//
// m_out = LN(m) over 256 ch;  z_out[i,j,:] = LN(z[i,j,:]) + (W.T[bin(d_ij)]+b)
//
// Roofline: ~605 MB streaming vs ~0.6 GFLOP -> HBM-bound (~26 us at 23.3 TB/s).
// No matmul exists in this op (one_hot@W is a 15-entry gather) -> no WMMA.
// gfx1250 paths used: wave32 shuffle reductions, NT b128 vector memory,
// global_prefetch_b8, and GLOBAL_LOAD_ASYNC_TO_LDS_B128 (+ s_wait_asynccnt)
// to stage the fused W/b table into LDS without a VGPR round-trip.
// ---------------------------------------------------------------------------

typedef float v4f __attribute__((ext_vector_type(4)));

#define N_RES 768
#define CM    256
#define CZ    128
#define NBINS 15
#define TBL_FLOATS 2048          // 15*128 = 1920 padded to 2048 (8 KB)
#define TBL_BYTES  (TBL_FLOATS * 4)
#define PAIRS_PER_BLOCK 128      // 8 waves x 16 rows
#define ROWS_PER_WAVE   16

// 5-step wave32 butterfly reduction (warpSize == 32 on gfx1250)
__device__ __forceinline__ float waveReduceSum(float v) {
#pragma unroll
  for (int off = 16; off > 0; off >>= 1)
    v += __shfl_xor(v, off, 32);
  return v;
}

// Low 32 bits of a generic pointer to __shared__ = byte offset in LDS space.
__device__ __forceinline__ unsigned ldsOffset(const void* p) {
  return (unsigned)(unsigned long long)p;
}

// ---------------------------------------------------------------------------
// Kernel 1: m_out = LayerNorm(m) over 256 channels.
// One wave32 per row; each lane owns 8 floats (two float4 slots).
// ---------------------------------------------------------------------------
__global__ void __launch_bounds__(256)
ln_m_kernel(const float* __restrict__ m,
            const float* __restrict__ gamma,
            const float* __restrict__ beta,
            float* __restrict__ out) {
  const int lane = threadIdx.x & 31;
  const int wave = threadIdx.x >> 5;
  const int row  = blockIdx.x * 8 + wave;
  if (row >= N_RES) return;

  const float* rp = m + (size_t)row * CM;
  const int c0 = lane * 4;
  const int c1 = 128 + lane * 4;

  v4f a  = *(const v4f*)(rp + c0);
  v4f b2 = *(const v4f*)(rp + c1);

  float s  = (a.x + a.y) + (a.z + a.w) + (b2.x + b2.y) + (b2.z + b2.w);
  float s2 = a.x * a.x + a.y * a.y + a.z * a.z + a.w * a.w
           + b2.x * b2.x + b2.y * b2.y + b2.z * b2.z + b2.w * b2.w;
  s  = waveReduceSum(s);
  s2 = waveReduceSum(s2);

  const float mu  = s * (1.0f / CM);
  const float var = fmaxf(s2 * (1.0f / CM) - mu * mu, 0.0f);
  const float rs  = __frsqrt_rn(var + 1e-5f);

  v4f g0  = *(const v4f*)(gamma + c0);
  v4f g1  = *(const v4f*)(gamma + c1);
  v4f be0 = *(const v4f*)(beta + c0);
  v4f be1 = *(const v4f*)(beta + c1);

  v4f o0 = (a  - mu) * rs * g0 + be0;
  v4f o1 = (b2 - mu) * rs * g1 + be1;

  float* op = out + (size_t)row * CM;
  *(v4f*)(op + c0) = o0;
  *(v4f*)(op + c1) = o1;
}

// ---------------------------------------------------------------------------
// Kernel A: build fused table once in global scratch:
//   tbl[k*128 + c] = W[c*15 + k] + b[c]   (contiguous -> async-copyable)
// ---------------------------------------------------------------------------
__global__ void __launch_bounds__(256)
build_table_kernel(const float* __restrict__ W,
                   const float* __restrict__ bvec,
                   float* __restrict__ tbl) {
  for (int idx = threadIdx.x; idx < TBL_FLOATS; idx += 256) {
    if (idx < NBINS * CZ) {
      const int k = idx >> 7;
      const int c = idx & (CZ - 1);
      tbl[idx] = W[c * NBINS + k] + bvec[c];
    } else {
      tbl[idx] = 0.0f;   // padding (never indexed: k*128+c <= 1916)
    }
  }
}

// ---------------------------------------------------------------------------
// Shared main loop: each wave streams ROWS_PER_WAVE consecutive (i,j) rows.
// Bins: linspace(3.25, 20.75, 15), step 1.25 -> nearest = round((d-3.25)*0.8)
// ---------------------------------------------------------------------------
__device__ __forceinline__ void z_main_loop(const float* __restrict__ z,
                                            const float* __restrict__ x,
                                            const float* ldsW,
                                            const float* __restrict__ gamma,
                                            const float* __restrict__ beta,
                                            float* __restrict__ out) {
  const int lane = threadIdx.x & 31;
  const int wave = threadIdx.x >> 5;
  const int c0   = lane * 4;

  const v4f g  = *(const v4f*)(gamma + c0);   // hoisted: constant per lane
  const v4f be = *(const v4f*)(beta + c0);

  int pair = blockIdx.x * PAIRS_PER_BLOCK + wave * ROWS_PER_WAVE;
  int i = pair / N_RES;
  int j = pair - i * N_RES;

  const float* zp = z   + (size_t)pair * CZ + c0;
  float*       op = out + (size_t)pair * CZ + c0;

#pragma unroll 2
  for (int it = 0; it < ROWS_PER_WAVE; ++it) {
    // Pull the row 4 iterations ahead (gfx1250 global_prefetch_b8; speculative,
    // OOB translation failures are silently dropped).
    __builtin_prefetch(zp + 4 * CZ, 0, 3);

    // Streaming NT 128-bit load: 512B per wave in one instruction.
    v4f zv = __builtin_nontemporal_load((const v4f*)zp);

    // LayerNorm statistics over 128 channels via wave32 butterfly.
    float s  = (zv.x + zv.y) + (zv.z + zv.w);
    float s2 = zv.x * zv.x + zv.y * zv.y + zv.z * zv.z + zv.w * zv.w;
    s  = waveReduceSum(s);
    s2 = waveReduceSum(s2);
    const float mu  = s * (1.0f / CZ);
    const float var = fmaxf(s2 * (1.0f / CZ) - mu * mu, 0.0f);
    const float rs  = __frsqrt_rn(var + 1e-5f);

    // Distance -> nearest bin (uniform per wave; x is 9 KB, lives in L0/L2).
    const float dx = x[i * 3 + 0] - x[j * 3 + 0];
    const float dy = x[i * 3 + 1] - x[j * 3 + 1];
    const float dz = x[i * 3 + 2] - x[j * 3 + 2];
    const float d  = sqrtf(dx * dx + dy * dy + dz * dz);
    int k = (int)floorf((d - 3.25f) * 0.8f + 0.5f);
    k = min(max(k, 0), NBINS - 1);

    // Fused embedding lookup from LDS (conflict-free ds_load_b128).
    v4f e = *(const v4f*)(ldsW + k * CZ + c0);

    v4f o = e + (zv - mu) * rs * g + be;
    __builtin_nontemporal_store(o, (v4f*)op);

    zp += CZ;
    op += CZ;
    if (++j == N_RES) { j = 0; ++i; }   // incremental i,j: no per-row division
  }
}

// ---------------------------------------------------------------------------
// Kernel 2a (preferred): table arrives in LDS via async copy from scratch.
// 256 threads x 2 x b128 = 8 KB; ASYNCcnt tracked, then workgroup barrier.
// ---------------------------------------------------------------------------
__global__ void __launch_bounds__(256)
z_fused_async_kernel(const float* __restrict__ z,
                     const float* __restrict__ x,
                     const float* __restrict__ tbl,
                     const float* __restrict__ gamma,
                     const float* __restrict__ beta,
                     float* __restrict__ out) {
  __shared__ float ldsW[TBL_FLOATS];

  {
    const unsigned tid  = threadIdx.x;
    const unsigned byt  = tid * 16u;                 // 16 B per lane per copy
    const unsigned l0   = ldsOffset(&ldsW[0]) + byt;
    const unsigned long long base = (unsigned long long)tbl;
    // Each wave issues two GLOBAL_LOAD_ASYNC_TO_LDS_B128 (512 B each):
    // 8 waves x 2 x 512 B = 8192 B = whole table. GVS mode: saddr + voffset.
    asm volatile("global_load_async_to_lds_b128 %0, %1, %2"
                 :: "v"(l0), "v"(byt), "s"(base) : "memory");
    asm volatile("global_load_async_to_lds_b128 %0, %1, %2"
                 :: "v"(l0 + 4096u), "v"(byt + 4096u), "s"(base) : "memory");
    asm volatile("s_wait_asynccnt 0x0" ::: "memory");
  }
  __syncthreads();

  z_main_loop(z, x, ldsW, gamma, beta, out);
}

// ---------------------------------------------------------------------------
// Kernel 2b (fallback if scratch too small): build table in LDS per block.
// ---------------------------------------------------------------------------
__global__ void __launch_bounds__(256)
z_fused_local_kernel(const float* __restrict__ z,
                     const float* __restrict__ x,
                     const float* __restrict__ W,
                     const float* __restrict__ bvec,
                     const float* __restrict__ gamma,
                     const float* __restrict__ beta,
                     float* __restrict__ out) {
  __shared__ float ldsW[TBL_FLOATS];

  for (int idx = threadIdx.x; idx < NBINS * CZ; idx += 256) {
    const int k = idx >> 7;
    const int c = idx & (CZ - 1);
    ldsW[idx] = W[c * NBINS + k] + bvec[c];
  }
  __syncthreads();

  z_main_loop(z, x, ldsW, gamma, beta, out);
}

// ---------------------------------------------------------------------------
// Launch
// ---------------------------------------------------------------------------
extern "C" void kernel_launch(void* const* d_in, const int* in_sizes, int n_in,
                              void* d_out, int out_size, void* d_ws, size_t ws_size,
                              hipStream_t stream) {
  (void)in_sizes; (void)n_in; (void)out_size;

  const float* m       = (const float*)d_in[0];
  const float* z       = (const float*)d_in[1];
  const float* x       = (const float*)d_in[2];
  const float* W       = (const float*)d_in[3];
  const float* bvec    = (const float*)d_in[4];
  const float* gamma_m = (const float*)d_in[5];
  const float* beta_m  = (const float*)d_in[6];
  const float* gamma_z = (const float*)d_in[7];
  const float* beta_z  = (const float*)d_in[8];

  float* m_out = (float*)d_out;                   // 768*256 floats
  float* z_out = m_out + (size_t)N_RES * CM;      // 768*768*128 floats

  // m layernorm: 768 rows, 8 rows per 256-thread block -> 96 blocks.
  ln_m_kernel<<<dim3(N_RES / 8), dim3(256), 0, stream>>>(m, gamma_m, beta_m, m_out);

  const int zBlocks = (N_RES * N_RES) / PAIRS_PER_BLOCK;   // 4608

  if (ws_size >= (size_t)TBL_BYTES && d_ws != nullptr) {
    float* tbl = (float*)d_ws;
    build_table_kernel<<<dim3(1), dim3(256), 0, stream>>>(W, bvec, tbl);
    z_fused_async_kernel<<<dim3(zBlocks), dim3(256), 0, stream>>>(
        z, x, tbl, gamma_z, beta_z, z_out);
  } else {
    z_fused_local_kernel<<<dim3(zBlocks), dim3(256), 0, stream>>>(
        z, x, W, bvec, gamma_z, beta_z, z_out);
  }
}